// IIRFilterbank_56092272886024
// MI455X (gfx1250) — compile-verified
//
#include <hip/hip_runtime.h>

typedef float v2f __attribute__((ext_vector_type(2)));
typedef float v8f __attribute__((ext_vector_type(8)));
typedef int   v4i __attribute__((vector_size(16)));

#define T_LEN   65536
#define NBATCH  128
#define NFILT   4
#define CHUNK   512
#define NCH_LG2 7
#define NCH     (T_LEN / CHUNK)   // 128 chunks
#define NBT     (NBATCH / 16)     // 8 batch tiles of 16
#define ACH_SQ  5                 // A^16 squared 5x -> A^512 = A^CHUNK

// workspace layout (floats)
#define TBL_F   512              // per-filter table block
#define H_OFF   0                // 16x16 Toeplitz block-response
#define OBS_OFF 256              // 16x4  observability rows e0^T A^i
#define WC_OFF  320              // 4x16  Wc[:,t] = A^(15-t) g
#define A16_OFF 384              // 4x4   A^16
#define ACH_OFF 400              // 4x4   A^CHUNK
#define E_OFF   (NFILT * TBL_F)                       // 2048
#define E_SZ    (NBATCH * NFILT * NCH * 4)            // 262144
#define S_OFF   (E_OFF + E_SZ)

#if defined(__has_builtin)
#if __has_builtin(__builtin_amdgcn_global_load_async_to_lds_b128) && \
    __has_builtin(__builtin_amdgcn_s_wait_asynccnt)
#define USE_ASYNC 1
#endif
#endif
#ifndef USE_ASYNC
#define USE_ASYNC 0
#endif

#if USE_ASYNC
typedef __attribute__((address_space(1))) v4i g_v4i;   // global int4
typedef __attribute__((address_space(3))) v4i l_v4i;   // LDS int4
#endif

// ---------------- kernel 0: per-filter state-space tables -----------------
__global__ void k_tables(const float* __restrict__ b, const float* __restrict__ a,
                         float* __restrict__ ws) {
  int f = threadIdx.x;
  if (f >= NFILT) return;
  float an[5], bn[5];
  float a0 = a[f * 5];
  for (int k = 0; k < 5; ++k) { an[k] = a[f * 5 + k] / a0; bn[k] = b[f * 5 + k] / a0; }
  float b0 = bn[0];
  float g[4];
  for (int i = 0; i < 4; ++i) g[i] = bn[i + 1] - an[i + 1] * b0;
  float A[4][4];
  for (int i = 0; i < 4; ++i)
    for (int j = 0; j < 4; ++j)
      A[i][j] = ((j == i + 1) ? 1.f : 0.f) - an[i + 1] * ((j == 0) ? 1.f : 0.f);

  float* tb = ws + f * TBL_F;

  // Markov parameters h[0..15] and Toeplitz H
  float h[16]; h[0] = b0;
  {
    float w[4] = {g[0], g[1], g[2], g[3]};
    for (int k = 1; k < 16; ++k) {
      h[k] = w[0];
      float w2[4];
      for (int i = 0; i < 4; ++i) { float s = 0.f; for (int j = 0; j < 4; ++j) s += A[i][j] * w[j]; w2[i] = s; }
      for (int i = 0; i < 4; ++i) w[i] = w2[i];
    }
  }
  for (int m = 0; m < 16; ++m)
    for (int k = 0; k < 16; ++k)
      tb[H_OFF + m * 16 + k] = (k > m) ? 0.f : h[m - k];

  // Obs rows: e0^T A^i
  {
    float r[4] = {1.f, 0.f, 0.f, 0.f};
    for (int i = 0; i < 16; ++i) {
      for (int j = 0; j < 4; ++j) tb[OBS_OFF + i * 4 + j] = r[j];
      float r2[4];
      for (int j = 0; j < 4; ++j) { float s = 0.f; for (int k = 0; k < 4; ++k) s += r[k] * A[k][j]; r2[j] = s; }
      for (int j = 0; j < 4; ++j) r[j] = r2[j];
    }
  }
  // Wc[:,t] = A^(15-t) g
  {
    float u[4] = {g[0], g[1], g[2], g[3]};
    for (int t = 15; t >= 0; --t) {
      for (int j = 0; j < 4; ++j) tb[WC_OFF + j * 16 + t] = u[j];
      float u2[4];
      for (int i = 0; i < 4; ++i) { float s = 0.f; for (int k = 0; k < 4; ++k) s += A[i][k] * u[k]; u2[i] = s; }
      for (int i = 0; i < 4; ++i) u[i] = u2[i];
    }
  }
  // A^16 and A^CHUNK (repeated squaring)
  float M[4][4];
  for (int i = 0; i < 4; ++i) for (int j = 0; j < 4; ++j) M[i][j] = A[i][j];
  for (int rep = 0; rep < 4; ++rep) {          // A -> A^16
    float M2[4][4];
    for (int i = 0; i < 4; ++i) for (int j = 0; j < 4; ++j) {
      float s = 0.f; for (int k = 0; k < 4; ++k) s += M[i][k] * M[k][j]; M2[i][j] = s;
    }
    for (int i = 0; i < 4; ++i) for (int j = 0; j < 4; ++j) M[i][j] = M2[i][j];
  }
  for (int i = 0; i < 4; ++i) for (int j = 0; j < 4; ++j) tb[A16_OFF + i * 4 + j] = M[i][j];
  for (int rep = 0; rep < ACH_SQ; ++rep) {     // A^16 -> A^CHUNK
    float M2[4][4];
    for (int i = 0; i < 4; ++i) for (int j = 0; j < 4; ++j) {
      float s = 0.f; for (int k = 0; k < 4; ++k) s += M[i][k] * M[k][j]; M2[i][j] = s;
    }
    for (int i = 0; i < 4; ++i) for (int j = 0; j < 4; ++j) M[i][j] = M2[i][j];
  }
  for (int i = 0; i < 4; ++i) for (int j = 0; j < 4; ++j) tb[ACH_OFF + i * 4 + j] = M[i][j];
}

// -------- kernel A: zero-state end state of each (seq, chunk) -------------
__global__ __launch_bounds__(256) void k_endstate(const float* __restrict__ x,
                                                  const float* __restrict__ b,
                                                  const float* __restrict__ a,
                                                  float* __restrict__ ws) {
  int gid = blockIdx.x * 256 + threadIdx.x;     // NBATCH*NFILT*NCH
  int c     = gid & (NCH - 1);
  int rest  = gid >> NCH_LG2;
  int f     = rest & 3;
  int batch = rest >> 2;
  float a0  = a[f * 5];
  float an1 = a[f * 5 + 1] / a0, an2 = a[f * 5 + 2] / a0, an3 = a[f * 5 + 3] / a0, an4 = a[f * 5 + 4] / a0;
  float bn0 = b[f * 5 + 0] / a0, bn1 = b[f * 5 + 1] / a0, bn2 = b[f * 5 + 2] / a0,
        bn3 = b[f * 5 + 3] / a0, bn4 = b[f * 5 + 4] / a0;
  const float4* xp = (const float4*)(x + ((size_t)batch * NFILT + f) * T_LEN + (size_t)c * CHUNK);
  float z0 = 0.f, z1 = 0.f, z2 = 0.f, z3 = 0.f;
#define IIR_STEP(xx) { float y = bn0 * (xx) + z0;              \
    z0 = z1 + bn1 * (xx) - an1 * y;                            \
    z1 = z2 + bn2 * (xx) - an2 * y;                            \
    z2 = z3 + bn3 * (xx) - an3 * y;                            \
    z3 =      bn4 * (xx) - an4 * y; }
  for (int i = 0; i < CHUNK / 4; ++i) {
    float4 v = xp[i];
    IIR_STEP(v.x) IIR_STEP(v.y) IIR_STEP(v.z) IIR_STEP(v.w)
  }
#undef IIR_STEP
  float* e = ws + E_OFF + ((size_t)(batch * NFILT + f) * NCH + c) * 4;
  e[0] = z0; e[1] = z1; e[2] = z2; e[3] = z3;
}

// -------- kernel B: cross-chunk scan  S_{c+1} = A^CHUNK S_c + e_c ---------
__global__ __launch_bounds__(256) void k_scan(float* __restrict__ ws) {
  int gid = blockIdx.x * 256 + threadIdx.x;
  if (gid >= NBATCH * NFILT) return;
  int f = gid & 3, batch = gid >> 2;
  const float* M = ws + f * TBL_F + ACH_OFF;
  float m00=M[0],m01=M[1],m02=M[2],m03=M[3], m10=M[4],m11=M[5],m12=M[6],m13=M[7],
        m20=M[8],m21=M[9],m22=M[10],m23=M[11], m30=M[12],m31=M[13],m32=M[14],m33=M[15];
  const float* e  = ws + E_OFF + (size_t)(batch * NFILT + f) * NCH * 4;
  float*       si = ws + S_OFF + (size_t)(batch * NFILT + f) * NCH * 4;
  float s0 = 0.f, s1 = 0.f, s2 = 0.f, s3 = 0.f;
  for (int c = 0; c < NCH; ++c) {
    si[c * 4 + 0] = s0; si[c * 4 + 1] = s1; si[c * 4 + 2] = s2; si[c * 4 + 3] = s3;
    float t0 = m00*s0 + m01*s1 + m02*s2 + m03*s3 + e[c * 4 + 0];
    float t1 = m10*s0 + m11*s1 + m12*s2 + m13*s3 + e[c * 4 + 1];
    float t2 = m20*s0 + m21*s1 + m22*s2 + m23*s3 + e[c * 4 + 2];
    float t3 = m30*s0 + m31*s1 + m32*s2 + m33*s3 + e[c * 4 + 3];
    s0 = t0; s1 = t1; s2 = t2; s3 = t3;
  }
}

// -------- kernel C: WMMA block filter, one wave per (f, batch-tile, chunk) ---
__global__ __launch_bounds__(32) void k_main(const float* __restrict__ x,
                                             float* __restrict__ out,
                                             const float* __restrict__ ws) {
  __shared__ float xs[2][16 * 68];   // double-buffered input macro-tile
  __shared__ float ys[16 * 68];
  int bid = blockIdx.x;
  int c  = bid % NCH;
  int bt = (bid / NCH) % NBT;
  int f  = bid / (NCH * NBT);
  int l   = threadIdx.x;
  int col = l & 15;          // N column (batch within tile)
  int hf  = l >> 4;          // wave half
  const float* tb = ws + f * TBL_F;
  int m = col;               // A-operand row M = lane%16

  // constant A-operands (16x4 layout: v0 = K 2*hf, v1 = K 2*hf+1)
  v2f HA[4], PA[4], OBSA, QA;
  for (int c4 = 0; c4 < 4; ++c4) {
    HA[c4].x = tb[H_OFF + m * 16 + 4 * c4 + 2 * hf];
    HA[c4].y = tb[H_OFF + m * 16 + 4 * c4 + 2 * hf + 1];
    PA[c4].x = (m < 4) ? tb[WC_OFF + m * 16 + 4 * c4 + 2 * hf]     : 0.f;
    PA[c4].y = (m < 4) ? tb[WC_OFF + m * 16 + 4 * c4 + 2 * hf + 1] : 0.f;
  }
  OBSA.x = tb[OBS_OFF + m * 4 + 2 * hf];
  OBSA.y = tb[OBS_OFF + m * 4 + 2 * hf + 1];
  QA.x = (m < 4) ? tb[A16_OFF + m * 4 + 2 * hf]     : 0.f;
  QA.y = (m < 4) ? tb[A16_OFF + m * 4 + 2 * hf + 1] : 0.f;

  // incoming chunk state for this lane's column (B-layout pair)
  int batchcol = bt * 16 + col;
  const float* si = ws + S_OFF + ((size_t)(batchcol * NFILT + f) * NCH + c) * 4;
  float bs0 = si[2 * hf], bs1 = si[2 * hf + 1];

  // coalesced staging addresses: 2 lanes per batch row, 32 floats each
  int rowio = l >> 1;
  int h32   = (l & 1) * 32;
  const float* xbase = x   + ((size_t)(bt * 16 + rowio) * NFILT + f) * T_LEN;
  float*       obase = out + ((size_t)(bt * 16 + rowio) * NFILT + f) * T_LEN;
  const int nmacro = CHUNK / 64;     // 8 macro-tiles of 64 timesteps

#if USE_ASYNC
  // async DMA prefetch: global -> LDS without the VGPR round trip (ASYNCcnt)
  {
    const float* gsrc = xbase + c * CHUNK + h32;
    float*       ldst = &xs[0][rowio * 68 + h32];
    #pragma unroll
    for (int q = 0; q < 8; ++q)
      __builtin_amdgcn_global_load_async_to_lds_b128(
          (g_v4i*)(gsrc + q * 4), (l_v4i*)(ldst + q * 4), 0, 0);
  }
#endif

  for (int mi = 0; mi < nmacro; ++mi) {
    int buf = mi & 1;
    int t0m = c * CHUNK + mi * 64;
#if USE_ASYNC
    if (mi + 1 < nmacro) {   // prefetch next macro-tile into the other buffer
      const float* gsrc = xbase + t0m + 64 + h32;
      float*       ldst = &xs[buf ^ 1][rowio * 68 + h32];
      #pragma unroll
      for (int q = 0; q < 8; ++q)
        __builtin_amdgcn_global_load_async_to_lds_b128(
            (g_v4i*)(gsrc + q * 4), (l_v4i*)(ldst + q * 4), 0, 0);
      __builtin_amdgcn_s_wait_asynccnt(8);  // loads retire in order: current tile done
    } else {
      __builtin_amdgcn_s_wait_asynccnt(0);
    }
    __syncthreads();
#else
    for (int q = 0; q < 8; ++q)
      *(float4*)&xs[buf][rowio * 68 + h32 + q * 4] = *(const float4*)(xbase + t0m + h32 + q * 4);
    __syncthreads();
#endif

    for (int ib = 0; ib < 4; ++ib) {
      v2f bst; bst.x = bs0; bst.y = bs1;
      v8f d   = {0.f,0.f,0.f,0.f,0.f,0.f,0.f,0.f};
      v8f dsx = {0.f,0.f,0.f,0.f,0.f,0.f,0.f,0.f};
      // Y  += Obs * S ;  S' += A16 * S
      d   = __builtin_amdgcn_wmma_f32_16x16x4_f32(false, OBSA, false, bst, (short)0, d,   false, false);
      dsx = __builtin_amdgcn_wmma_f32_16x16x4_f32(false, QA,   false, bst, (short)0, dsx, false, false);
      // Y += H * X ; S' += Wc * X   (K=16 in four K=4 chunks)
      for (int c4 = 0; c4 < 4; ++c4) {
        float2 bxv = *(const float2*)&xs[buf][col * 68 + ib * 16 + 4 * c4 + 2 * hf];
        v2f bx; bx.x = bxv.x; bx.y = bxv.y;
        d   = __builtin_amdgcn_wmma_f32_16x16x4_f32(false, HA[c4], false, bx, (short)0, d,   false, false);
        dsx = __builtin_amdgcn_wmma_f32_16x16x4_f32(false, PA[c4], false, bx, (short)0, dsx, false, false);
      }
      // next-block state lives in rows 0..3 of dsx (lanes 0-15); replicate to half 1
      float u2 = __shfl(dsx[2], col, 32);
      float u3 = __shfl(dsx[3], col, 32);
      bs0 = hf ? u2 : dsx[0];
      bs1 = hf ? u3 : dsx[1];
      // D rows -> LDS: lane holds Y[v + 8*hf][col] in component v
      int yb = col * 68 + ib * 16 + 8 * hf;
      for (int v = 0; v < 8; v += 2) {
        float2 p; p.x = d[v]; p.y = d[v + 1];
        *(float2*)&ys[yb + v] = p;
      }
    }
    __syncthreads();
    // stage out (128B contiguous per lane)
    for (int q = 0; q < 8; ++q)
      *(float4*)(obase + t0m + h32 + q * 4) = *(float4*)&ys[rowio * 68 + h32 + q * 4];
    __syncthreads();
  }
}

extern "C" void kernel_launch(void* const* d_in, const int* in_sizes, int n_in,
                              void* d_out, int out_size, void* d_ws, size_t ws_size,
                              hipStream_t stream) {
  const float* x = (const float*)d_in[0];
  const float* b = (const float*)d_in[1];
  const float* a = (const float*)d_in[2];
  float* out = (float*)d_out;
  float* ws  = (float*)d_ws;
  hipLaunchKernelGGL(k_tables,   dim3(1),                           dim3(32),  0, stream, b, a, ws);
  hipLaunchKernelGGL(k_endstate, dim3(NBATCH * NFILT * NCH / 256),  dim3(256), 0, stream, x, b, a, ws);
  hipLaunchKernelGGL(k_scan,     dim3(2),                           dim3(256), 0, stream, ws);
  hipLaunchKernelGGL(k_main,     dim3(NFILT * NBT * NCH),           dim3(32),  0, stream, x, out, ws);
}